// ExponentialSmoothing_7438883356763
// MI455X (gfx1250) — compile-verified
//
#include <hip/hip_runtime.h>
#include <math.h>

// ExponentialSmoothing on MI455X (gfx1250):
//   y[n] = w^{n+1} * v0 + sum_{m<=n} (1-w) w^{n-m} x[m],  w = sigmoid(sw[h])
// Memory-bound (268 MB min traffic @ 23.3 TB/s ~= 11.5us). 3-pass chunked scan,
// chunk C=16; intra-chunk triangular operator applied with V_WMMA_F32_16X16X4_F32.
// Pass-3 output stores are non-temporal so `values` (134 MB) stays resident in
// the 192 MB L2 between pass 1 and pass 3.

typedef __attribute__((ext_vector_type(2))) float v2f;
typedef __attribute__((ext_vector_type(8))) float v8f;

constexpr int kB  = 16;
constexpr int kT  = 4096;
constexpr int kH  = 8;
constexpr int kD  = 64;
constexpr int kC  = 16;          // chunk length (matches WMMA M/K tile)
constexpr int kNC = kT / kC;     // 256 chunks

__device__ __forceinline__ float sigmoidf_(float x) {
    return 1.0f / (1.0f + __expf(-x));
}

// ---------------- Pass 1: per-chunk weighted reduction --------------------
// contrib[b][h][c][d] = (1-w) * sum_{m=0..15} w^{15-m} * x[b, c*16+m, h, d]
__global__ void __launch_bounds__(256)
ema_chunk_contrib(const float* __restrict__ x,
                  const float* __restrict__ sw,
                  float* __restrict__ contrib) {
    int gid = blockIdx.x * 256 + threadIdx.x;        // [0, kB*kH*kNC*kD) = 2^21
    int d = gid & (kD - 1);
    int c = (gid >> 6) & (kNC - 1);
    int h = (gid >> 14) & (kH - 1);
    int b = gid >> 17;

    float w = sigmoidf_(sw[h]);
    const size_t rowStride = (size_t)kH * kD;        // 512 floats between t-steps
    const float* p = x + (((size_t)(b * kT + c * kC) * kH + h) * kD + d);

    float acc = 0.0f, wk = 1.0f;                     // wk = w^{15-m}, m descending
    #pragma unroll
    for (int m = kC - 1; m >= 0; --m) {
        acc = fmaf(wk, p[(size_t)m * rowStride], acc);
        wk *= w;
    }
    contrib[gid] = (1.0f - w) * acc;
}

// ---------------- Pass 2: tiny serial scan over chunk states --------------
// carry[b][h][c][d] = EMA state just BEFORE chunk c (carry[.,0]=v0)
// s_{c+1} = w^16 * s_c + contrib_c
__global__ void ema_chunk_scan(const float* __restrict__ contrib,
                               const float* __restrict__ sw,
                               const float* __restrict__ v0,
                               float* __restrict__ carry) {
    int d  = threadIdx.x;                            // 64 lanes of d
    int bh = blockIdx.x;                             // b*8 + h, 128 blocks
    int h  = bh & (kH - 1);

    float w = sigmoidf_(sw[h]);
    float w16 = 1.0f;
    #pragma unroll
    for (int i = 0; i < kC; ++i) w16 *= w;

    float s = v0[h * kD + d];
    size_t base = (size_t)bh * kNC * kD + d;
    for (int c = 0; c < kNC; ++c) {
        carry[base + (size_t)c * kD] = s;
        s = fmaf(w16, s, contrib[base + (size_t)c * kD]);
    }
}

// ---------------- Pass 3: intra-chunk apply via WMMA ----------------------
// Y(16x64) = L(16x16) * X(16x64) + diag(w^{j+1}) * carry,  L[n][m]=(n>=m)(1-w)w^{n-m}
// One wave per (b,h,chunk); 8 waves per block; EXEC all-ones (no divergence).
// All register arrays use COMPILE-TIME indices only (no cndmask ladders);
// lane-dependent powers of w come from binary decomposition of the lane id.
__global__ void __launch_bounds__(256)
ema_chunk_apply(const float* __restrict__ x,
                const float* __restrict__ sw,
                const float* __restrict__ carry,
                float* __restrict__ out) {
    int wave = threadIdx.x >> 5;
    int lane = threadIdx.x & 31;
    int idx  = blockIdx.x * 8 + wave;                // chunk id in [0, 32768)
    int c = idx & (kNC - 1);
    int h = (idx >> 8) & (kH - 1);
    int b = idx >> 11;

    float w   = sigmoidf_(sw[h]);
    float omw = 1.0f - w;

    // wp[i] = w^i, built and consumed with constant indices only.
    float wp[kC + 1];
    wp[0] = 1.0f;
    #pragma unroll
    for (int i = 1; i <= kC; ++i) wp[i] = wp[i - 1] * w;

    // invp[i] = w^-i, constant-indexed.
    float invw = 1.0f / w;
    float invp[kC];
    invp[0] = 1.0f;
    #pragma unroll
    for (int i = 1; i < kC; ++i) invp[i] = invp[i - 1] * invw;

    int  hiHalf = (lane >= 16) ? 1 : 0;
    int  l15    = lane & 15;

    // per-lane w^l15 via binary decomposition: 3 muls + selects, no table lookup
    float pw2 = w * w, pw4 = pw2 * pw2, pw8 = pw4 * pw4;
    float pl = ((l15 & 1) ? w   : 1.0f)
             * ((l15 & 2) ? pw2 : 1.0f)
             * ((l15 & 4) ? pw4 : 1.0f)
             * ((l15 & 8) ? pw8 : 1.0f);
    float plw = omw * pl;                            // (1-w) * w^l15

    // A (16x4 f32 per K-step): lanes 0-15 hold K={4k,4k+1}, lanes 16-31 K={4k+2,4k+3},
    // row M = lane&15 (ISA 7.12.2).  L[r][m] = (r>=m) * (1-w) * w^r * w^-m.
    v2f a[4];
    #pragma unroll
    for (int k = 0; k < 4; ++k) {
        // hiHalf-dependent pick between two compile-time-indexed entries
        float i0 = hiHalf ? invp[4 * k + 2] : invp[4 * k + 0];
        float i1 = hiHalf ? invp[4 * k + 3] : invp[4 * k + 1];
        int   m0 = 4 * k + 2 * hiHalf;
        float a0 = (l15 >= m0)     ? plw * i0 : 0.0f;
        float a1 = (l15 >= m0 + 1) ? plw * i1 : 0.0f;
        a[k] = (v2f){a0, a1};
    }

    const size_t rowStride = (size_t)kH * kD;
    const float* xbase = x   + (((size_t)(b * kT + c * kC) * kH + h) * kD);
    float*       obase = out + (((size_t)(b * kT + c * kC) * kH + h) * kD);
    const float* cbase = carry + (((size_t)(b * kH + h) * kNC + c) * kD);

    float hi8 = hiHalf ? wp[8] : 1.0f;               // w^{8*hiHalf}

    #pragma unroll
    for (int n = 0; n < 4; ++n) {                    // N-tiles over d
        int col = n * 16 + l15;
        float cvh = cbase[col] * hi8;                // carry * w^{8*hiHalf}

        // C/D layout: VGPR r -> row M = r + 8*hiHalf, col = lane&15
        v8f acc;
        #pragma unroll
        for (int r = 0; r < 8; ++r)
            acc[r] = wp[r + 1] * cvh;                // w^{row+1} * carry, const idx

        #pragma unroll
        for (int k = 0; k < 4; ++k) {                // K-steps (4 t-rows each)
            int krow = 4 * k + 2 * hiHalf;           // B 4x16: rows striped like A
            v2f bm;
            bm.x = __builtin_nontemporal_load(&xbase[(size_t)krow * rowStride + col]);
            bm.y = __builtin_nontemporal_load(&xbase[(size_t)(krow + 1) * rowStride + col]);
            // (neg_a, A, neg_b, B, c_mod, C, reuse_a, reuse_b)
            acc = __builtin_amdgcn_wmma_f32_16x16x4_f32(
                false, a[k], false, bm, (short)0, acc, false, false);
        }

        #pragma unroll
        for (int r = 0; r < 8; ++r) {
            int row = r + 8 * hiHalf;
            // NT store: don't let output evict the L2-resident input stream
            __builtin_nontemporal_store(acc[r],
                &obase[(size_t)row * rowStride + col]);
        }
    }
}

extern "C" void kernel_launch(void* const* d_in, const int* in_sizes, int n_in,
                              void* d_out, int out_size, void* d_ws, size_t ws_size,
                              hipStream_t stream) {
    const float* values = (const float*)d_in[0];   // [16,4096,8,64] f32
    const float* sw     = (const float*)d_in[1];   // [8,1] f32
    const float* v0     = (const float*)d_in[2];   // [1,1,8,64] f32
    float* out = (float*)d_out;

    const size_t nChunkElems = (size_t)kB * kH * kNC * kD;   // 2,097,152
    float* contrib = (float*)d_ws;                            // 8 MB
    float* carry   = contrib + nChunkElems;                   // 8 MB

    ema_chunk_contrib<<<(int)(nChunkElems / 256), 256, 0, stream>>>(values, sw, contrib);
    ema_chunk_scan<<<kB * kH, kD, 0, stream>>>(contrib, sw, v0, carry);
    ema_chunk_apply<<<(kB * kH * kNC) / 8, 256, 0, stream>>>(values, sw, carry, out);
}